// ContactRecommenderGNN_14628658610510
// MI455X (gfx1250) — compile-verified
//
#include <hip/hip_runtime.h>

// CDNA5 / gfx1250. wave32. f32 WMMA 16x16x4.
typedef __attribute__((ext_vector_type(2))) float v2f;
typedef __attribute__((ext_vector_type(8))) float v8f;

#define BN_EPS 1e-5f

// Native f32 atomic add (agent scope, relaxed) -> global_atomic_add_f32,
// avoiding any CAS-loop fallback.
__device__ __forceinline__ void atomic_fadd(float* p, float v) {
    __hip_atomic_fetch_add(p, v, __ATOMIC_RELAXED, __HIP_MEMORY_SCOPE_AGENT);
}

// ---------------------------------------------------------------------------
// Degree kernels (graph is shared across all 3 layers -> compute once)
// ---------------------------------------------------------------------------
__global__ void deg_accum_kernel(const int* __restrict__ dst, float* __restrict__ deg, int E) {
    int e = blockIdx.x * blockDim.x + threadIdx.x;
    if (e < E) atomic_fadd(&deg[dst[e]], 1.0f);
}

__global__ void invdeg_kernel(const float* __restrict__ deg, float* __restrict__ invdeg, int N) {
    int i = blockIdx.x * blockDim.x + threadIdx.x;
    if (i < N) invdeg[i] = 1.0f / fmaxf(deg[i], 1.0f);
}

// ---------------------------------------------------------------------------
// Scatter: agg[dst, k] += x[src, k]   (one thread per (edge, feature))
// Feature index = tid % D so x reads and atomic targets coalesce per wave.
// ---------------------------------------------------------------------------
template<int D>
__global__ void scatter_add_kernel(const float* __restrict__ x,
                                   const int* __restrict__ src,
                                   const int* __restrict__ dst,
                                   float* __restrict__ agg,
                                   int E) {
    int gid = blockIdx.x * blockDim.x + threadIdx.x;
    int e = gid / D;
    int k = gid - e * D;
    if (e >= E) return;
    int s = src[e];
    int d = dst[e];
    atomic_fadd(&agg[d * D + k], x[s * D + k]);
}

// ---------------------------------------------------------------------------
// WMMA f32 GEMM:  out[N,DOUT] = (agg * invdeg_row) @ Wl + xin @ Wr + bias
//   One block = one 16-row stripe; wave w owns columns [16w, 16w+16).
//   A fragment (16x4 f32, ISA layout): lanes 0-15 hold K=k0,k0+1 (vgpr0,1),
//   lanes 16-31 hold K=k0+2,k0+3.  B symmetric.  C/D: vgpr r = row r (+8 for
//   upper lane half), col = lane&15.
//   K, DOUT compile-time: K=64 -> unconditional (merged b64) loads, fully
//   unrolled; K=3 -> address-clamped loads + v_cndmask zeroing, no EXEC
//   branching.  N % 16 == 0 (50000) so EXEC is all-ones around every WMMA.
//   WITH_STATS: fuse BatchNorm sum/sumsq reduction into the epilogue
//   (lane l and l^16 hold the same output column -> one shfl_xor combine).
// ---------------------------------------------------------------------------
template<int K, int DOUT>
__device__ __forceinline__ v8f mac_panel(const float* __restrict__ A,
                                         const float* __restrict__ W,
                                         int mrow, int col, float scale, int kb,
                                         v8f acc) {
    constexpr int KP = (K + 3) & ~3;   // K padded to multiple of 4
    #pragma unroll
    for (int k0 = 0; k0 < KP; k0 += 4) {
        int ka = k0 + kb;              // this lane's first K index in the chunk
        v2f a, b;
        if constexpr ((K & 3) == 0) {
            a.x = A[mrow * K + ka]     * scale;
            a.y = A[mrow * K + ka + 1] * scale;
            b.x = W[ka * DOUT + col];
            b.y = W[(ka + 1) * DOUT + col];
        } else {
            int kc0 = (ka     < K) ? ka     : K - 1;
            int kc1 = (ka + 1 < K) ? ka + 1 : K - 1;
            float a0 = A[mrow * K + kc0];
            float a1 = A[mrow * K + kc1];
            float b0 = W[kc0 * DOUT + col];
            float b1 = W[kc1 * DOUT + col];
            a.x = (ka     < K) ? a0 * scale : 0.0f;
            a.y = (ka + 1 < K) ? a1 * scale : 0.0f;
            b.x = (ka     < K) ? b0 : 0.0f;
            b.y = (ka + 1 < K) ? b1 : 0.0f;
        }
        acc = __builtin_amdgcn_wmma_f32_16x16x4_f32(
            /*neg_a=*/false, a, /*neg_b=*/false, b,
            /*c_mod=*/(short)0, acc, /*reuse_a=*/false, /*reuse_b=*/false);
    }
    return acc;
}

template<int K, int DOUT, bool WITH_STATS>
__global__ void sage_gemm_wmma_kernel(const float* __restrict__ agg,
                                      const float* __restrict__ xin,
                                      const float* __restrict__ invdeg,
                                      const float* __restrict__ Wl,
                                      const float* __restrict__ Wr,
                                      const float* __restrict__ bias,
                                      float* __restrict__ out,
                                      float* __restrict__ stats) {
    int wave = threadIdx.x >> 5;
    int lane = threadIdx.x & 31;
    int r0 = blockIdx.x * 16;          // row block
    int c0 = wave * 16;                // column block (one wave per 16 cols)
    int mrow = r0 + (lane & 15);       // A-fragment row for this lane
    int col  = c0 + (lane & 15);       // B/C-fragment column for this lane
    int kb   = (lane >> 4) * 2;        // lane-half K offset within a 4-chunk

    v8f acc = {};
    float s = invdeg[mrow];
    acc = mac_panel<K, DOUT>(agg, Wl, mrow, col, s,    kb, acc);  // lin_l(mean)
    acc = mac_panel<K, DOUT>(xin, Wr, mrow, col, 1.0f, kb, acc);  // lin_r(x)

    float bi = bias[col];
    int orow = r0 + 8 * (lane >> 4);   // C/D layout: upper lane half = rows +8
    #pragma unroll
    for (int r = 0; r < 8; ++r)
        out[(orow + r) * DOUT + col] = acc[r] + bi;

    if constexpr (WITH_STATS) {
        // BN partial reduction fused into the epilogue: this lane holds 8 of
        // the 16 rows of column `col`; the other 8 live in lane^16.
        float psum = 0.0f, psq = 0.0f;
        #pragma unroll
        for (int r = 0; r < 8; ++r) {
            float v = acc[r] + bi;
            psum += v;
            psq  += v * v;
        }
        psum += __shfl_xor(psum, 16, 32);
        psq  += __shfl_xor(psq,  16, 32);
        if (lane < 16) {
            atomic_fadd(&stats[col],        psum);
            atomic_fadd(&stats[DOUT + col], psq);
        }
    }
}

// ---------------------------------------------------------------------------
// BatchNorm finalize + fused scale/shift/ReLU apply (H = 64)
// ---------------------------------------------------------------------------
__global__ void bn_finalize_kernel(const float* __restrict__ stats,
                                   const float* __restrict__ gamma,
                                   const float* __restrict__ beta,
                                   float* __restrict__ scsh, int N) {
    int f = threadIdx.x;
    if (f < 64) {
        float inv_n = 1.0f / (float)N;
        float mu  = stats[f] * inv_n;
        float var = stats[64 + f] * inv_n - mu * mu;   // biased variance
        float rstd = rsqrtf(var + BN_EPS);
        float scale = gamma[f] * rstd;
        scsh[f]      = scale;
        scsh[64 + f] = beta[f] - mu * scale;
    }
}

__global__ void bn_apply_relu_kernel(float4* __restrict__ h,
                                     const float* __restrict__ scsh, int Nvec) {
    int gid = blockIdx.x * blockDim.x + threadIdx.x;   // one float4 per thread
    if (gid < Nvec) {
        int f4 = (gid & 15) * 4;                       // 64 feats = 16 float4s
        float4 v = h[gid];
        v.x = fmaxf(0.0f, v.x * scsh[f4 + 0] + scsh[64 + f4 + 0]);
        v.y = fmaxf(0.0f, v.y * scsh[f4 + 1] + scsh[64 + f4 + 1]);
        v.z = fmaxf(0.0f, v.z * scsh[f4 + 2] + scsh[64 + f4 + 2]);
        v.w = fmaxf(0.0f, v.w * scsh[f4 + 3] + scsh[64 + f4 + 3]);
        h[gid] = v;
    }
}

// ---------------------------------------------------------------------------
// Launch
// ---------------------------------------------------------------------------
extern "C" void kernel_launch(void* const* d_in, const int* in_sizes, int n_in,
                              void* d_out, int out_size, void* d_ws, size_t ws_size,
                              hipStream_t stream) {
    const float* x   = (const float*)d_in[0];
    const int*   ei  = (const int*)d_in[1];
    const float* Wl0 = (const float*)d_in[2];
    const float* bl0 = (const float*)d_in[3];
    const float* Wr0 = (const float*)d_in[4];
    const float* g0  = (const float*)d_in[5];
    const float* b0  = (const float*)d_in[6];
    const float* Wl1 = (const float*)d_in[7];
    const float* bl1 = (const float*)d_in[8];
    const float* Wr1 = (const float*)d_in[9];
    const float* g1  = (const float*)d_in[10];
    const float* b1  = (const float*)d_in[11];
    const float* Wl2 = (const float*)d_in[12];
    const float* bl2 = (const float*)d_in[13];
    const float* Wr2 = (const float*)d_in[14];

    const int N = in_sizes[0] / 3;     // 50000 (divisible by 16)
    const int E = in_sizes[1] / 2;     // 800000
    const int* src = ei;
    const int* dst = ei + E;

    // workspace carve-up (f32)
    float* agg    = (float*)d_ws;               // N*64
    float* h0     = agg    + (size_t)N * 64;    // N*64
    float* h1     = h0     + (size_t)N * 64;    // N*64
    float* deg    = h1     + (size_t)N * 64;    // N
    float* invdeg = deg    + N;                 // N
    float* stats  = invdeg + N;                 // 128 (sum, sumsq)
    float* scsh   = stats  + 128;               // 128 (scale, shift)

    const int TB = 256;
    const int nv4 = N * 16;            // N*64 floats = N*16 float4s

    // ---- degree (once) ----
    hipMemsetAsync(deg, 0, (size_t)N * sizeof(float), stream);
    deg_accum_kernel<<<(E + TB - 1) / TB, TB, 0, stream>>>(dst, deg, E);
    invdeg_kernel<<<(N + TB - 1) / TB, TB, 0, stream>>>(deg, invdeg, N);

    // ---- layer 0: SAGE(3 -> 64) + BN + ReLU ----
    hipMemsetAsync(agg, 0, (size_t)N * 64 * sizeof(float), stream);
    {
        long long tot = (long long)E * 3;
        scatter_add_kernel<3><<<(int)((tot + TB - 1) / TB), TB, 0, stream>>>(x, src, dst, agg, E);
    }
    hipMemsetAsync(stats, 0, 128 * sizeof(float), stream);
    sage_gemm_wmma_kernel<3, 64, true><<<N / 16, 32 * (64 / 16), 0, stream>>>(
        agg, x, invdeg, Wl0, Wr0, bl0, h0, stats);
    bn_finalize_kernel<<<1, 64, 0, stream>>>(stats, g0, b0, scsh, N);
    bn_apply_relu_kernel<<<(nv4 + TB - 1) / TB, TB, 0, stream>>>((float4*)h0, scsh, nv4);

    // ---- layer 1: SAGE(64 -> 64) + BN + ReLU ----
    hipMemsetAsync(agg, 0, (size_t)N * 64 * sizeof(float), stream);
    {
        long long tot = (long long)E * 64;
        scatter_add_kernel<64><<<(int)((tot + TB - 1) / TB), TB, 0, stream>>>(h0, src, dst, agg, E);
    }
    hipMemsetAsync(stats, 0, 128 * sizeof(float), stream);
    sage_gemm_wmma_kernel<64, 64, true><<<N / 16, 32 * (64 / 16), 0, stream>>>(
        agg, h0, invdeg, Wl1, Wr1, bl1, h1, stats);
    bn_finalize_kernel<<<1, 64, 0, stream>>>(stats, g1, b1, scsh, N);
    bn_apply_relu_kernel<<<(nv4 + TB - 1) / TB, TB, 0, stream>>>((float4*)h1, scsh, nv4);

    // ---- layer 2: SAGE(64 -> 128), no BN/ReLU, straight to d_out ----
    hipMemsetAsync(agg, 0, (size_t)N * 64 * sizeof(float), stream);
    {
        long long tot = (long long)E * 64;
        scatter_add_kernel<64><<<(int)((tot + TB - 1) / TB), TB, 0, stream>>>(h1, src, dst, agg, E);
    }
    sage_gemm_wmma_kernel<64, 128, false><<<N / 16, 32 * (128 / 16), 0, stream>>>(
        agg, h1, invdeg, Wl2, Wr2, bl2, (float*)d_out, nullptr);
}